// UnionLayer_2327872274610
// MI455X (gfx1250) — compile-verified
//
#include <hip/hip_runtime.h>

// Problem shape (fixed by the reference)
#define B_ 8
#define N_ 64
#define L_ 64
#define C_ 768
#define S_ (N_ * L_)   // 4096 tokens per document
#define ROWS_PER_BLOCK 4

// ---------------------------------------------------------------------------
// Kernel 2 (placed first so the disasm snippet shows the async-LDS path):
// the ~200 MB data movement. 256-thread blocks (8 wave32); each 64-thread
// group copies one 768-float output row. Valid rows ride the CDNA5 async
// data mover: GLOBAL_LOAD_ASYNC_TO_LDS_B128 -> s_wait_asynccnt ->
// GLOBAL_STORE_ASYNC_FROM_LDS_B128 (ASYNCcnt-tracked, no VGPR round-trip).
// Pad rows are zero-filled with b128 stores (d_out is poisoned once, so every
// element must be written each call). The map[] branch is wave-uniform, so
// EXEC is all-1s around the async ops as the ISA requires.
// ---------------------------------------------------------------------------
__global__ void kie_pack_kernel(const float* __restrict__ src,   // (B,S,C)
                                const int*   __restrict__ map,   // (B,S)
                                float*       __restrict__ dst)   // (B,S,C)
{
    __shared__ float buf[ROWS_PER_BLOCK][C_];     // 12 KB staging
    const int g   = threadIdx.x >> 6;             // row group 0..3
    const int t   = threadIdx.x & 63;             // lane within group
    const int row = blockIdx.x * ROWS_PER_BLOCK + g;   // 0 .. B*S-1
    const int b   = row >> 12;                    // row / 4096
    const int m   = map[row];                     // uniform per 64-thread group

    float* __restrict__ drow = dst + (size_t)row * C_;

    if (m >= 0) {
        const float* __restrict__ srow = src + ((size_t)b * S_ + m) * C_;
        // Wave-relative LDS byte offset: low 32 bits of the generic address
        // of a __shared__ object are the LDS offset (aperture mapping).
        unsigned lds_base = (unsigned)(size_t)(const void*)&buf[g][0];

        // 64 lanes x 16 B x 3 rounds = 3072 B = one row
        #pragma unroll
        for (int r = 0; r < 3; ++r) {
            const int   e    = t + r * 64;                 // float4 index
            unsigned    loff = lds_base + (unsigned)(e * 16);
            const void* gp   = (const void*)(srow + e * 4);
            asm volatile("global_load_async_to_lds_b128 %0, %1, off"
                         :: "v"(loff), "v"(gp) : "memory");
        }
        asm volatile("s_wait_asynccnt 0x0" ::: "memory");   // LDS data landed
        #pragma unroll
        for (int r = 0; r < 3; ++r) {
            const int e    = t + r * 64;
            unsigned  loff = lds_base + (unsigned)(e * 16);
            void*     gp   = (void*)(drow + e * 4);
            asm volatile("global_store_async_from_lds_b128 %0, %1, off"
                         :: "v"(gp), "v"(loff) : "memory");
        }
        asm volatile("s_wait_asynccnt 0x0" ::: "memory");   // stores drained
    } else {
        const float4 z = make_float4(0.f, 0.f, 0.f, 0.f);
        float4* d4 = (float4*)drow;
        #pragma unroll
        for (int r = 0; r < 3; ++r) d4[t + r * 64] = z;
    }
}

// ---------------------------------------------------------------------------
// Kernel 1: per-line valid lengths, prefix sum, dest->src row map, masks.
// One block per batch. 8 blocks x 256 threads. Negligible work.
// ---------------------------------------------------------------------------
__global__ void kie_setup_kernel(const int* __restrict__ texts,      // (B,N,L) int32
                                 const int* __restrict__ end_val_p,  // scalar
                                 int*       __restrict__ map,        // ws: (B,S)
                                 float*     __restrict__ out_newmask,// (B,S)
                                 float*     __restrict__ out_mask)   // (B,N*L)
{
    __shared__ int s_vlen[N_];
    __shared__ int s_off[N_ + 1];
    __shared__ int s_map[S_];        // 16 KB of the 320 KB WGP LDS

    const int b   = blockIdx.x;
    const int tid = threadIdx.x;
    const int ev  = end_val_p[0];

    // vlen[n] = first index where token == end_val, else 0 (argmax semantics)
    if (tid < N_) {
        const int n = tid;
        const int* line = texts + ((size_t)b * N_ + n) * L_;
        int vlen = 0;
        #pragma clang loop unroll(disable)
        for (int l = 0; l < L_; ++l) {
            if (line[l] == ev) { vlen = l; break; }
        }
        s_vlen[n] = vlen;
    }
    __syncthreads();

    // Exclusive prefix sum over 64 lines (serial: 64 adds, negligible)
    if (tid == 0) {
        int acc = 0;
        #pragma clang loop unroll(disable)
        for (int n = 0; n < N_; ++n) { s_off[n] = acc; acc += s_vlen[n]; }
        s_off[N_] = acc;
    }
    // Init map to -1 (pad rows)
    for (int j = tid; j < S_; j += blockDim.x) s_map[j] = -1;
    __syncthreads();

    // Fill valid dest slots: dest off[n]+l  <-  src flat row n*L + l
    if (tid < N_) {
        const int n    = tid;
        const int off  = s_off[n];
        const int vlen = s_vlen[n];
        #pragma clang loop unroll(disable)
        for (int l = 0; l < vlen; ++l) s_map[off + l] = n * L_ + l;
    }
    __syncthreads();

    const int doclen = s_off[N_];
    for (int j = tid; j < S_; j += blockDim.x) {
        map[(size_t)b * S_ + j]         = s_map[j];
        out_newmask[(size_t)b * S_ + j] = (j < doclen) ? 1.0f : 0.0f;
    }
    for (int idx = tid; idx < N_ * L_; idx += blockDim.x) {
        const int n = idx / L_;
        const int l = idx % L_;
        out_mask[(size_t)b * (N_ * L_) + idx] = (l < s_vlen[n]) ? 1.0f : 0.0f;
    }
}

// ---------------------------------------------------------------------------
// Host-side launcher (graph-capture safe: only kernel launches on stream)
// ---------------------------------------------------------------------------
extern "C" void kernel_launch(void* const* d_in, const int* in_sizes, int n_in,
                              void* d_out, int out_size, void* d_ws, size_t ws_size,
                              hipStream_t stream) {
    (void)in_sizes; (void)n_in; (void)out_size; (void)ws_size;

    const float* kie     = (const float*)d_in[0];   // (B,N,L,C) f32
    const int*   texts   = (const int*)d_in[1];     // (B,N,L) int
    const int*   end_val = (const int*)d_in[2];     // scalar int

    float* out_logits  = (float*)d_out;                          // (B,S,C)
    float* out_newmask = out_logits + (size_t)B_ * S_ * C_;      // (B,S)
    float* out_mask    = out_newmask + (size_t)B_ * S_;          // (B,N*L)

    int* map = (int*)d_ws;                                       // 128 KB scratch

    kie_setup_kernel<<<B_, 256, 0, stream>>>(texts, end_val, map,
                                             out_newmask, out_mask);
    kie_pack_kernel<<<(B_ * S_) / ROWS_PER_BLOCK, 256, 0, stream>>>(kie, map, out_logits);
}